// QuantumCircuitLayer_944892805728
// MI455X (gfx1250) — compile-verified
//
#include <hip/hip_runtime.h>
#include <math.h>

typedef __attribute__((ext_vector_type(2))) float v2f;
typedef __attribute__((ext_vector_type(8))) float v8f;

#define NQ 12
#define DEPTH 3
#define DIM 4096   // 2^12
#define BLK 256    // 8 wave32s

__device__ __forceinline__ v8f wmma4(v2f a, v2f b, v8f c) {
    // D(16x16,f32) = A(16x4,f32) x B(4x16,f32) + C
    return __builtin_amdgcn_wmma_f32_16x16x4_f32(
        false, a, false, b, (short)0, c, false, false);
}

// state index from (group g, row r = 4 group bits, col c = other 8 bits)
__device__ __forceinline__ int sidx(int g, int r, int c) {
    if (g == 0) return (c << 4) | r;
    if (g == 1) return (c & 15) | (r << 4) | ((c >> 4) << 8);
    return c | (r << 8);
}

__global__ void __launch_bounds__(BLK)
qcircuit_kernel(const float* __restrict__ inputs,  // [B,12]
                const float* __restrict__ theta,   // [3,12,3]
                float* __restrict__ out)           // [B,12]
{
    __shared__ float sRe[DIM];
    __shared__ float sIm[DIM];
    __shared__ float Kre[256];
    __shared__ float Kim[256];
    __shared__ float g2Re[NQ][4];
    __shared__ float g2Im[NQ][4];
    __shared__ float cw[NQ];
    __shared__ float sw[NQ];
    __shared__ float evAcc[NQ];

    const int tid  = threadIdx.x;
    const int b    = blockIdx.x;
    const int lane = tid & 31;
    const int wave = tid >> 5;
    const int mrow = lane & 15;   // A row / B,D column-within-tile
    const int half = lane >> 4;

    // ---- init |0...0> and per-sample RY cos/sin ----
    for (int i = tid; i < DIM; i += BLK) { sRe[i] = (i == 0) ? 1.0f : 0.0f; sIm[i] = 0.0f; }
    if (tid < NQ) {
        float a = 0.5f * inputs[b * NQ + tid];
        cw[tid] = cosf(a);
        sw[tid] = sinf(a);
    }
    __syncthreads();

    // ---------- phase 0: RY encoding (real 16x16 per group), then 3 Rot layers ----------
    for (int phase = 0; phase <= DEPTH; ++phase) {
        if (phase >= 1) {
            int l = phase - 1;
            if (tid < NQ) {  // per-wire Rot 2x2 (complex), shared theta
                const float* tp = theta + (l * NQ + tid) * 3;
                float phi = tp[0], th = tp[1], om = tp[2];
                float ct = cosf(0.5f * th), st = sinf(0.5f * th);
                float aP = 0.5f * (phi + om), aM = 0.5f * (phi - om);
                float cP = cosf(aP), sP = sinf(aP);
                float cM = cosf(aM), sM = sinf(aM);
                g2Re[tid][0] =  ct * cP; g2Im[tid][0] = -ct * sP;  // g00
                g2Re[tid][1] = -st * cM; g2Im[tid][1] = -st * sM;  // g01
                g2Re[tid][2] =  st * cM; g2Im[tid][2] = -st * sM;  // g10
                g2Re[tid][3] =  ct * cP; g2Im[tid][3] =  ct * sP;  // g11
            }
            __syncthreads();
        }

        for (int g = 0; g < 3; ++g) {
            // ---- build 16x16 group gate K (Kronecker product of four 2x2) ----
            {
                int m = tid >> 4, r = tid & 15;
                if (phase == 0) {
                    float pr = 1.0f;
#pragma unroll
                    for (int j = 0; j < 4; ++j) {
                        int w = 4 * g + j;
                        int mj = (m >> j) & 1, rj = (r >> j) & 1;
                        float e = (mj == rj) ? cw[w] : (mj ? sw[w] : -sw[w]);
                        pr *= e;
                    }
                    Kre[tid] = pr; Kim[tid] = 0.0f;
                } else {
                    float pr = 1.0f, pi = 0.0f;
#pragma unroll
                    for (int j = 0; j < 4; ++j) {
                        int w = 4 * g + j;
                        int mj = (m >> j) & 1, rj = (r >> j) & 1;
                        float er = g2Re[w][mj * 2 + rj];
                        float ei = g2Im[w][mj * 2 + rj];
                        float nr = pr * er - pi * ei;
                        float ni = pr * ei + pi * er;
                        pr = nr; pi = ni;
                    }
                    Kre[tid] = pr; Kim[tid] = pi;
                }
            }
            __syncthreads();

            // ---- apply K to state via WMMA: newS(16x256) = K(16x16) * S(16x256) ----
            v2f aRe[4], aIm[4], aImN[4];
#pragma unroll
            for (int j = 0; j < 4; ++j) {
                int k0 = 4 * j + 2 * half;
                aRe[j].x = Kre[mrow * 16 + k0]; aRe[j].y = Kre[mrow * 16 + k0 + 1];
                aIm[j].x = Kim[mrow * 16 + k0]; aIm[j].y = Kim[mrow * 16 + k0 + 1];
                aImN[j] = -aIm[j];
            }
#pragma unroll
            for (int tt = 0; tt < 2; ++tt) {        // 16 column tiles over 8 waves
                int t   = wave * 2 + tt;
                int col = t * 16 + mrow;
                v2f bRe[4], bIm[4];
#pragma unroll
                for (int j = 0; j < 4; ++j) {
                    int r0 = 4 * j + 2 * half;
                    int i0 = sidx(g, r0,     col);
                    int i1 = sidx(g, r0 + 1, col);
                    bRe[j].x = sRe[i0]; bRe[j].y = sRe[i1];
                    bIm[j].x = sIm[i0]; bIm[j].y = sIm[i1];
                }
                v8f accR = {}; v8f accI = {};
#pragma unroll
                for (int j = 0; j < 4; ++j) accR = wmma4(aRe[j],  bRe[j], accR); // +Kre*Sre
#pragma unroll
                for (int j = 0; j < 4; ++j) accR = wmma4(aImN[j], bIm[j], accR); // -Kim*Sim
#pragma unroll
                for (int j = 0; j < 4; ++j) accI = wmma4(aRe[j],  bIm[j], accI); // +Kre*Sim
#pragma unroll
                for (int j = 0; j < 4; ++j) accI = wmma4(aIm[j],  bRe[j], accI); // +Kim*Sre
#pragma unroll
                for (int p = 0; p < 8; ++p) {
                    int m  = p + 8 * half;
                    int io = sidx(g, m, col);
                    sRe[io] = accR[p];
                    sIm[io] = accI[p];
                }
            }
            __syncthreads();
        }

        // ---- CNOT ring after each Rot layer ----
        if (phase >= 1) {
            for (int e = 0; e < NQ; ++e) {
                int ctl = (e < NQ - 1) ? e : NQ - 1;
                int tgt = (e < NQ - 1) ? e + 1 : 0;
                for (int p = tid; p < (DIM >> 2); p += BLK) {  // 1024 swap pairs
                    int i = 1 << ctl;      // ctl bit = 1, tgt bit = 0
                    int pp = p;
                    for (int bpos = 0; bpos < NQ; ++bpos) {
                        if (bpos == ctl || bpos == tgt) continue;
                        i |= (pp & 1) << bpos;
                        pp >>= 1;
                    }
                    int j2 = i | (1 << tgt);
                    float tr = sRe[i], ti = sIm[i];
                    sRe[i] = sRe[j2]; sIm[i] = sIm[j2];
                    sRe[j2] = tr;     sIm[j2] = ti;
                }
                __syncthreads();
            }
        }
    }

    // ---------- measurement: ev[w] = sum_i |amp_i|^2 * (1 - 2*bit_w(i)) ----------
    if (tid < NQ) evAcc[tid] = 0.0f;
    float lacc[NQ];
#pragma unroll
    for (int w = 0; w < NQ; ++w) lacc[w] = 0.0f;
    for (int i = tid; i < DIM; i += BLK) {
        float pr = sRe[i], pi = sIm[i];
        float pb = pr * pr + pi * pi;
#pragma unroll
        for (int w = 0; w < NQ; ++w)
            lacc[w] += ((i >> w) & 1) ? -pb : pb;
    }
    __syncthreads();
#pragma unroll
    for (int w = 0; w < NQ; ++w) atomicAdd(&evAcc[w], lacc[w]);
    __syncthreads();
    if (tid < NQ) out[b * NQ + tid] = evAcc[tid];
}

extern "C" void kernel_launch(void* const* d_in, const int* in_sizes, int n_in,
                              void* d_out, int out_size, void* d_ws, size_t ws_size,
                              hipStream_t stream) {
    const float* inputs = (const float*)d_in[0];   // [B,12] f32
    const float* theta  = (const float*)d_in[1];   // [3,12,3] f32
    float* out = (float*)d_out;                    // [B,12] f32
    int B = in_sizes[0] / NQ;
    qcircuit_kernel<<<B, BLK, 0, stream>>>(inputs, theta, out);
}